// PDE_DGM_BlackScholes_48902497632674
// MI455X (gfx1250) — compile-verified
//
#include <hip/hip_runtime.h>
#include <math.h>

// ---------------------------------------------------------------------------
// DGM Black-Scholes PDE residual, fully fused, second-order forward-mode AD.
// 12 propagated channels per row: [val, d/dt, d/dx1..d/dx5, d2/dx1^2..d2/dx5^2]
// Heavy math: 144 GEMMs [B x128]x[128x128] f32 -> V_WMMA_F32_16X16X4_F32.
// Z+G+R gates share one k-loop: each A-fragment DS load feeds 3 WMMAs.
// ---------------------------------------------------------------------------

#define B_TOT   32768
#define D_IN    5
#define HID     128
#define ROWS    16              // batch rows per workgroup (one WMMA M-tile)
#define HIDS    132             // padded LDS row stride (floats) - bank-conflict free
#define CH      (ROWS * HIDS)   // floats per channel buffer = 2112
#define NCH     12
#define KROWS   (D_IN + 1 + HID)  // 134 rows per gate weight
#define MU      0.05f
#define SIGMA   0.2f

typedef float v2f __attribute__((ext_vector_type(2)));
typedef float v8f __attribute__((ext_vector_type(8)));

__device__ __forceinline__ v8f wmma4(v2f a, v2f b, v8f c) {
  return __builtin_amdgcn_wmma_f32_16x16x4_f32(false, a, false, b, (short)0, c,
                                               false, false);
}

// Single-gate: D_c += A_c(16x128, LDS) x W_bot(128 x 16-col tile, L2), c=0..11.
__device__ __forceinline__ void wmma_gate(const float* A, const float* __restrict__ W,
                                          int lane, int n0, v8f acc[NCH]) {
  const int arow = lane & 15;
  const int koff = (lane >> 4) << 1;     // lanes 16-31 carry K+2,K+3
  const float* wp = W + (6 + koff) * HID + (n0 + arow);  // skip 6 tx rows
  const float* ap = A + arow * HIDS + koff;
  #pragma unroll 4
  for (int k = 0; k < 32; ++k) {
    v2f b;
    b.x = wp[k * (4 * HID)];
    b.y = wp[k * (4 * HID) + HID];
    #pragma unroll
    for (int c = 0; c < NCH; ++c) {
      v2f a = *(const v2f*)(ap + c * CH + k * 4);
      acc[c] = wmma4(a, b, acc[c]);
    }
  }
}

// Fused three-gate (shared A operand): each A-fragment DS load feeds 3 WMMAs.
__device__ __forceinline__ void wmma_gate3(const float* A,
                                           const float* __restrict__ W0,
                                           const float* __restrict__ W1,
                                           const float* __restrict__ W2,
                                           int lane, int n0,
                                           v8f acc0[NCH], v8f acc1[NCH],
                                           v8f acc2[NCH]) {
  const int arow = lane & 15;
  const int koff = (lane >> 4) << 1;
  const float* wp0 = W0 + (6 + koff) * HID + (n0 + arow);
  const float* wp1 = W1 + (6 + koff) * HID + (n0 + arow);
  const float* wp2 = W2 + (6 + koff) * HID + (n0 + arow);
  const float* ap  = A + arow * HIDS + koff;
  #pragma unroll 2
  for (int k = 0; k < 32; ++k) {
    v2f b0, b1, b2;
    b0.x = wp0[k * (4 * HID)];
    b0.y = wp0[k * (4 * HID) + HID];
    b1.x = wp1[k * (4 * HID)];
    b1.y = wp1[k * (4 * HID) + HID];
    b2.x = wp2[k * (4 * HID)];
    b2.y = wp2[k * (4 * HID) + HID];
    #pragma unroll
    for (int c = 0; c < NCH; ++c) {
      v2f a = *(const v2f*)(ap + c * CH + k * 4);
      acc0[c] = wmma4(a, b0, acc0[c]);
      acc1[c] = wmma4(a, b1, acc1[c]);
      acc2[c] = wmma4(a, b2, acc2[c]);
    }
  }
}

// Add tx @ W_top + bias to the value channel; first-derivative channels get
// the corresponding W_top row (d(tx)/dv is one-hot). Second derivs: nothing.
__device__ __forceinline__ void add_top(const float* __restrict__ W,
                                        const float* __restrict__ b,
                                        const float* txs,
                                        int lane, int n0, v8f acc[NCH]) {
  const int col = n0 + (lane & 15);
  const int rb  = (lane >> 4) << 3;
  float cw[6];
  #pragma unroll
  for (int j = 0; j < 6; ++j) cw[j] = W[j * HID + col];
  const float bb = b[col];
  #pragma unroll
  for (int i = 0; i < 8; ++i) {
    float a0 = bb;
    #pragma unroll
    for (int j = 0; j < 6; ++j) a0 += txs[(rb + i) * 6 + j] * cw[j];
    acc[0][i] += a0;
    #pragma unroll
    for (int v = 0; v < 6; ++v) acc[1 + v][i] += cw[v];
  }
}

// In-place tanh propagation through the 12-channel jet.
__device__ __forceinline__ void tanh_prop(float a[NCH]) {
  float y  = tanhf(a[0]);
  float yp = 1.f - y * y;
  float d1[6];
  #pragma unroll
  for (int v = 0; v < 6; ++v) d1[v] = a[1 + v];
  a[0] = y;
  #pragma unroll
  for (int v = 0; v < 6; ++v) a[1 + v] = yp * d1[v];
  #pragma unroll
  for (int q = 0; q < 5; ++q)
    a[7 + q] = yp * a[7 + q] - 2.f * y * yp * d1[1 + q] * d1[1 + q];
}

__global__ __launch_bounds__(256, 1)
void dgm_pde_kernel(const float* __restrict__ t,   const float* __restrict__ x,
                    const float* __restrict__ Win, const float* __restrict__ b_in,
                    const float* __restrict__ Wg,  const float* __restrict__ bg,
                    const float* __restrict__ Wout,const float* __restrict__ bout,
                    float* __restrict__ out) {
  extern __shared__ float smem[];
  float* S    = smem;                   // 12 * 2112 floats
  float* SR   = smem + NCH * CH;        // 12 * 2112 floats
  float* txs  = smem + 2 * NCH * CH;    // 16 * 6
  float* udot = txs + ROWS * 6;         // 12 * 16

  const int tid  = threadIdx.x;
  const int lane = tid & 31;
  const int wave = tid >> 5;
  const int n0   = wave * 16;           // this wave's N-tile
  const int r0   = blockIdx.x * ROWS;

  if (tid < ROWS) {
    txs[tid * 6 + 0] = t[r0 + tid];
    #pragma unroll
    for (int j = 0; j < D_IN; ++j) txs[tid * 6 + 1 + j] = x[(r0 + tid) * D_IN + j];
  }
  __syncthreads();

  // ---- input layer: S0 = tanh(tx @ Win + b_in), jet-propagated ----
  #pragma unroll
  for (int e = 0; e < 8; ++e) {
    int idx = tid * 8 + e;
    int row = idx >> 7, col = idx & 127;
    float d1[6], a0 = b_in[col];
    #pragma unroll
    for (int j = 0; j < 6; ++j) { d1[j] = Win[j * HID + col]; a0 += txs[row * 6 + j] * d1[j]; }
    float y = tanhf(a0), yp = 1.f - y * y;
    float* p = S + row * HIDS + col;
    p[0] = y;
    #pragma unroll
    for (int v = 0; v < 6; ++v) p[(1 + v) * CH] = yp * d1[v];
    #pragma unroll
    for (int q = 0; q < 5; ++q) p[(7 + q) * CH] = -2.f * y * yp * d1[1 + q] * d1[1 + q];
  }
  __syncthreads();

  const int col_l = n0 + (lane & 15);   // pointwise (row,col) per C-fragment slot
  const int rb    = (lane >> 4) << 3;

  for (int layer = 0; layer < 3; ++layer) {
    const float* Wl = Wg + (size_t)layer * 4 * KROWS * HID;
    const float* bl = bg + layer * 4 * HID;

    // ---- Phase A: Z+G+R fused over the S jet; SR = jet-product(S, tanh(R)) ----
    v8f accZ[NCH], accG[NCH], accR[NCH];
    #pragma unroll
    for (int c = 0; c < NCH; ++c) { accZ[c] = {}; accG[c] = {}; accR[c] = {}; }
    wmma_gate3(S, Wl + 0 * KROWS * HID, Wl + 1 * KROWS * HID, Wl + 2 * KROWS * HID,
               lane, n0, accZ, accG, accR);
    add_top(Wl + 0 * KROWS * HID, bl + 0 * HID, txs, lane, n0, accZ);
    add_top(Wl + 1 * KROWS * HID, bl + 1 * HID, txs, lane, n0, accG);
    add_top(Wl + 2 * KROWS * HID, bl + 2 * HID, txs, lane, n0, accR);

    #pragma unroll
    for (int i = 0; i < 8; ++i) {
      int row = rb + i;
      float r[NCH], s[NCH];
      #pragma unroll
      for (int c = 0; c < NCH; ++c) {
        r[c] = accR[c][i];
        s[c] = S[c * CH + row * HIDS + col_l];
      }
      tanh_prop(r);
      float sr[NCH];
      sr[0] = s[0] * r[0];
      #pragma unroll
      for (int v = 0; v < 6; ++v) sr[1 + v] = s[1 + v] * r[0] + s[0] * r[1 + v];
      #pragma unroll
      for (int q = 0; q < 5; ++q)
        sr[7 + q] = s[7 + q] * r[0] + 2.f * s[2 + q] * r[2 + q] + s[0] * r[7 + q];
      #pragma unroll
      for (int c = 0; c < NCH; ++c) SR[c * CH + row * HIDS + col_l] = sr[c];
    }
    __syncthreads();

    // ---- Phase B: H over the SR jet; S_new = (1-G)H + Z*S (jet rules) ----
    v8f accH[NCH];
    #pragma unroll
    for (int c = 0; c < NCH; ++c) accH[c] = {};
    wmma_gate(SR, Wl + 3 * KROWS * HID, lane, n0, accH);
    add_top(Wl + 3 * KROWS * HID, bl + 3 * HID, txs, lane, n0, accH);

    #pragma unroll
    for (int i = 0; i < 8; ++i) {
      int row = rb + i;
      float z[NCH], g[NCH], h[NCH], s[NCH];
      #pragma unroll
      for (int c = 0; c < NCH; ++c) {
        z[c] = accZ[c][i]; g[c] = accG[c][i]; h[c] = accH[c][i];
        s[c] = S[c * CH + row * HIDS + col_l];
      }
      tanh_prop(z); tanh_prop(g); tanh_prop(h);
      float omg = 1.f - g[0];
      accZ[0][i] = omg * h[0] + z[0] * s[0];   // reuse accZ to hold S_new jet
      #pragma unroll
      for (int v = 0; v < 6; ++v)
        accZ[1 + v][i] = -g[1 + v] * h[0] + omg * h[1 + v]
                       + z[1 + v] * s[0] + z[0] * s[1 + v];
      #pragma unroll
      for (int q = 0; q < 5; ++q)
        accZ[7 + q][i] = -g[7 + q] * h[0] - 2.f * g[2 + q] * h[2 + q] + omg * h[7 + q]
                       + z[7 + q] * s[0] + 2.f * z[2 + q] * s[2 + q] + z[0] * s[7 + q];
    }
    __syncthreads();                        // all waves done reading old S
    #pragma unroll
    for (int i = 0; i < 8; ++i) {
      int row = rb + i;
      #pragma unroll
      for (int c = 0; c < NCH; ++c) S[c * CH + row * HIDS + col_l] = accZ[c][i];
    }
    __syncthreads();
  }

  // ---- output layer: u_ch = S_ch @ Wout (+bout), then PDE residual ----
  if (tid < NCH * ROWS) {
    int ch = tid >> 4, row = tid & 15;
    const float* p = S + ch * CH + row * HIDS;
    float sum = (ch == 0) ? bout[0] : 0.f;
    for (int c2 = 0; c2 < HID; ++c2) sum += p[c2] * Wout[c2];
    udot[ch * 16 + row] = sum;
  }
  __syncthreads();
  if (tid < ROWS) {
    int row = tid;
    float u  = udot[row];
    float ut = udot[16 + row];
    float drift = 0.f, lap = 0.f;
    #pragma unroll
    for (int q = 0; q < 5; ++q) {
      drift += txs[row * 6 + 1 + q] * udot[(2 + q) * 16 + row];
      lap   += udot[(7 + q) * 16 + row];
    }
    out[r0 + row] = ut + MU * drift + 0.5f * SIGMA * SIGMA * lap - MU * u;
  }
}

extern "C" void kernel_launch(void* const* d_in, const int* in_sizes, int n_in,
                              void* d_out, int out_size, void* d_ws, size_t ws_size,
                              hipStream_t stream) {
  const float* t    = (const float*)d_in[0];
  const float* x    = (const float*)d_in[1];
  const float* Win  = (const float*)d_in[2];
  const float* b_in = (const float*)d_in[3];
  const float* Wg   = (const float*)d_in[4];
  const float* bg   = (const float*)d_in[5];
  const float* Wout = (const float*)d_in[6];
  const float* bout = (const float*)d_in[7];
  float* out = (float*)d_out;

  const size_t lds = (size_t)(2 * NCH * CH + ROWS * 6 + NCH * ROWS) * sizeof(float);
  (void)hipFuncSetAttribute((const void*)dgm_pde_kernel,
                            hipFuncAttributeMaxDynamicSharedMemorySize, (int)lds);
  hipLaunchKernelGGL(dgm_pde_kernel, dim3(B_TOT / ROWS), dim3(256), lds, stream,
                     t, x, Win, b_in, Wg, bg, Wout, bout, out);
}